// StaarModelTrainer_326417515371
// MI455X (gfx1250) — compile-verified
//
#include <hip/hip_runtime.h>
#include <hip/hip_bf16.h>
#include <math.h>

typedef float v2f __attribute__((ext_vector_type(2)));
typedef float v8f __attribute__((ext_vector_type(8)));

#define FEATURES   8
#define NB         4096
#define NT         512
#define NROWS      (NB * NT)          /* 2,097,152 rows of (8 x, 8 mean, 8 logvar) */
#define NSER       (NT + NB - 1)      /* 4607 series length */
#define NLAGS      50
#define NLL_BLOCKS 2048
#define GARCH_BLOCKS 18               /* ceil(4607/256) */
#define GARCH_WINDOW 320              /* beta^320 ~ 5e-23: below f32/f64 relevance */

#define OMEGA_F  0.05f
#define ALPHA_F  0.1f
#define BETA_F   0.85f
#define SIGMA0_F 1.0f                 /* omega/(1-alpha-beta) */
#define C0_F    (-0.98082925301172622f)  /* gammaln(2.5)-gammaln(2)-0.5*ln(4*pi) */
#define LOG_2PI_D 1.8378770664093453

/* workspace layout (floats) */
#define WS_NLL    0        /* [2048] per-block NLL partials                 */
#define WS_SER    2048     /* [4607] series                                 */
#define WS_MEAN   6656     /* [1]    series mean                            */
#define WS_CORR   6660     /* [51]   autocorr sums, corr[0] = variance      */
#define WS_GARCH  6720     /* [18]   per-block sum of log-prob              */

/* ---------------- dominant kernel: Gaussian NLL partial reduction -------- */
__global__ __launch_bounds__(256) void nll_partial_kernel(
    const float* __restrict__ x, const float* __restrict__ dec,
    float* __restrict__ partials) {
  const int total = NLL_BLOCKS * 256;
  int g = blockIdx.x * 256 + threadIdx.x;
  float acc = 0.0f;
  /* NROWS == 4 * total: uniform loop, EXEC stays all-ones (WMMA requirement) */
  for (int r = g; r < NROWS; r += total) {
    const float4* xr = (const float4*)(x + (size_t)r * 8);
    const float4* dr = (const float4*)(dec + (size_t)r * 16);
    float4 x0 = xr[0], x1 = xr[1];
    float4 m0 = dr[0], m1 = dr[1];
    float4 l0 = dr[2], l1 = dr[3];
    float d;
    d = x0.x - m0.x; acc += l0.x + d * d * __expf(-l0.x);
    d = x0.y - m0.y; acc += l0.y + d * d * __expf(-l0.y);
    d = x0.z - m0.z; acc += l0.z + d * d * __expf(-l0.z);
    d = x0.w - m0.w; acc += l0.w + d * d * __expf(-l0.w);
    d = x1.x - m1.x; acc += l1.x + d * d * __expf(-l1.x);
    d = x1.y - m1.y; acc += l1.y + d * d * __expf(-l1.y);
    d = x1.z - m1.z; acc += l1.z + d * d * __expf(-l1.z);
    d = x1.w - m1.w; acc += l1.w + d * d * __expf(-l1.w);
  }
  /* wave32 reduction via V_WMMA_F32_16X16X4_F32 with a ones B-matrix:
     A(16x4): lane l<16 -> A[l][0]=acc, lane l>=16 -> A[l-16][2]=acc, rest 0.
     D[m][n] = acc_m + acc_{m+16}. Sum of D's 8 VGPRs on a lane gives the
     half-row-set total; shfl_xor(16) merges the two halves. */
  v2f a;    a.x = acc;  a.y = 0.0f;
  v2f ones; ones.x = 1.0f; ones.y = 1.0f;
  v8f c = {};
  v8f dm = __builtin_amdgcn_wmma_f32_16x16x4_f32(
      false, a, false, ones, (short)0, c, false, false);
  float s8 = dm[0] + dm[1] + dm[2] + dm[3] + dm[4] + dm[5] + dm[6] + dm[7];
  float wavesum = s8 + __shfl_xor(s8, 16, 32);

  __shared__ float wsum[8];
  int lane = threadIdx.x & 31, wid = threadIdx.x >> 5;
  if (lane == 0) wsum[wid] = wavesum;
  __syncthreads();
  if (threadIdx.x == 0) {
    float bs = 0.0f;
    for (int i = 0; i < 8; ++i) bs += wsum[i];
    partials[blockIdx.x] = bs;
  }
}

/* ---------------- series = concat(y_pred[0,:,0], y_pred[1:,-1,0]) -------- */
__global__ __launch_bounds__(256) void build_series_kernel(
    const float* __restrict__ dec, float* __restrict__ s) {
  int i = blockIdx.x * 256 + threadIdx.x;
  if (i >= NSER) return;
  int idx;
  if (i < NT) idx = i * 16;                         /* b=0, t=i, f=0     */
  else { int b = i - (NT - 1); idx = (b * NT + (NT - 1)) * 16; } /* b,511,0 */
  s[i] = dec[idx];
}

/* ---------------- mean of series ---------------------------------------- */
__global__ __launch_bounds__(256) void series_mean_kernel(
    const float* __restrict__ s, float* __restrict__ meanout) {
  __shared__ float red[256];
  float acc = 0.0f;
  for (int i = threadIdx.x; i < NSER; i += 256) acc += s[i];
  red[threadIdx.x] = acc;
  __syncthreads();
  for (int st = 128; st > 0; st >>= 1) {
    if (threadIdx.x < st) red[threadIdx.x] += red[threadIdx.x + st];
    __syncthreads();
  }
  if (threadIdx.x == 0) meanout[0] = red[0] / (float)NSER;
}

/* ---------------- lag sums: corr[lag] = sum_t sc[t]*sc[t+lag] ------------ */
__global__ __launch_bounds__(256) void corr_kernel(
    const float* __restrict__ s, const float* __restrict__ meanp,
    float* __restrict__ corr) {
  int lag = blockIdx.x;          /* 0..50; corr[0] == variance */
  float mu = meanp[0];
  float acc = 0.0f;
  for (int t = threadIdx.x; t < NSER - lag; t += 256)
    acc += (s[t] - mu) * (s[t + lag] - mu);
  __shared__ float red[256];
  red[threadIdx.x] = acc;
  __syncthreads();
  for (int st = 128; st > 0; st >>= 1) {
    if (threadIdx.x < st) red[threadIdx.x] += red[threadIdx.x + st];
    __syncthreads();
  }
  if (threadIdx.x == 0) corr[lag] = red[0];
}

/* ---------------- GARCH NLL via windowed linear recurrence --------------- */
__global__ __launch_bounds__(256) void garch_partial_kernel(
    const float* __restrict__ s, const float* __restrict__ meanp,
    float* __restrict__ partials) {
  int t = blockIdx.x * 256 + threadIdx.x;
  float mu = meanp[0];
  float lp = 0.0f;
  if (t < NSER) {
    int j0 = t - (GARCH_WINDOW - 1);
    if (j0 < 0) j0 = 0;
    float cv = 0.0f, w = 1.0f;
    for (int j = t; j >= j0; --j) {        /* cv = sum beta^{t-j} (omega + alpha r2_j) */
      float r = s[j] - mu;
      cv += w * (OMEGA_F + ALPHA_F * r * r);
      w *= BETA_F;
    }
    if (j0 == 0) cv += w * SIGMA0_F;       /* w == beta^{t+1} here */
    float rt = s[t] - mu;
    float scale = sqrtf(cv * 0.5f);        /* (nu-2)/nu = 0.5 */
    float z = rt / scale;
    lp = C0_F - logf(scale) - 2.5f * log1pf(z * z * 0.25f);
  }
  __shared__ float red[256];
  red[threadIdx.x] = lp;
  __syncthreads();
  for (int st = 128; st > 0; st >>= 1) {
    if (threadIdx.x < st) red[threadIdx.x] += red[threadIdx.x + st];
    __syncthreads();
  }
  if (threadIdx.x == 0) partials[blockIdx.x] = red[0];
}

/* ---------------- deterministic finalize --------------------------------- */
__global__ __launch_bounds__(256) void finalize_kernel(
    const float* __restrict__ nllp, const float* __restrict__ corr,
    const float* __restrict__ garchp, float* __restrict__ out) {
  __shared__ double red[256];
  double acc = 0.0;
  for (int i = threadIdx.x; i < NLL_BLOCKS; i += 256) acc += (double)nllp[i];
  red[threadIdx.x] = acc;
  __syncthreads();
  for (int st = 128; st > 0; st >>= 1) {
    if (threadIdx.x < st) red[threadIdx.x] += red[threadIdx.x + st];
    __syncthreads();
  }
  if (threadIdx.x == 0) {
    double nll_sum = red[0];
    double nelem = (double)NROWS * (double)FEATURES;
    double recon = (0.5 / (double)NB) * (nll_sum + nelem * LOG_2PI_D);
    double var = (double)corr[0];
    double aclsum = 0.0;
    for (int l = 1; l <= NLAGS; ++l) {
      double a = (double)corr[l] / var;
      aclsum += a * a;
    }
    double acl = aclsum / (double)NLAGS;
    double gsum = 0.0;
    for (int i = 0; i < GARCH_BLOCKS; ++i) gsum += (double)garchp[i];
    double garch = -gsum;
    out[0] = (float)(recon + 10.0 * acl + 0.001 * garch);
  }
}

extern "C" void kernel_launch(void* const* d_in, const int* in_sizes, int n_in,
                              void* d_out, int out_size, void* d_ws, size_t ws_size,
                              hipStream_t stream) {
  const float* x   = (const float*)d_in[0];   /* x_true  (4096,512,8)  f32 */
  const float* dec = (const float*)d_in[1];   /* dec_out (4096,512,16) f32 */
  float* ws  = (float*)d_ws;
  float* out = (float*)d_out;

  nll_partial_kernel<<<NLL_BLOCKS, 256, 0, stream>>>(x, dec, ws + WS_NLL);
  build_series_kernel<<<GARCH_BLOCKS, 256, 0, stream>>>(dec, ws + WS_SER);
  series_mean_kernel<<<1, 256, 0, stream>>>(ws + WS_SER, ws + WS_MEAN);
  corr_kernel<<<NLAGS + 1, 256, 0, stream>>>(ws + WS_SER, ws + WS_MEAN, ws + WS_CORR);
  garch_partial_kernel<<<GARCH_BLOCKS, 256, 0, stream>>>(ws + WS_SER, ws + WS_MEAN, ws + WS_GARCH);
  finalize_kernel<<<1, 256, 0, stream>>>(ws + WS_NLL, ws + WS_CORR, ws + WS_GARCH, out);
}